// FBRNN_decoder_86904368267682
// MI455X (gfx1250) — compile-verified
//
#include <hip/hip_runtime.h>
#include <hip/hip_bf16.h>
#include <math.h>

// ---------------- sizes (fixed by the reference) ----------------
#define TT 512      // decoder steps
#define SS 512      // encoder length
#define HH 1024     // hidden
#define AA 256      // attention hidden
#define VV 50257    // vocab
#define NV_TILES 3142  // ceil(50257/16)

typedef __attribute__((ext_vector_type(2))) float v2f;
typedef __attribute__((ext_vector_type(8))) float v8f;

// ---------------- wave32 helpers ----------------
__device__ __forceinline__ float waveSum(float v) {
#pragma unroll
  for (int o = 16; o > 0; o >>= 1) v += __shfl_xor(v, o, 32);
  return v;
}
__device__ __forceinline__ float waveMax(float v) {
#pragma unroll
  for (int o = 16; o > 0; o >>= 1) v = fmaxf(v, __shfl_xor(v, o, 32));
  return v;
}
__device__ __forceinline__ float sigmf(float x) { return 1.0f / (1.0f + __expf(-x)); }
__device__ __forceinline__ float dot4(float4 a, float4 b) {
  return a.x * b.x + a.y * b.y + a.z * b.z + a.w * b.w;
}

// ---------------- K0: token embedding gather (b128) ----------------
__global__ __launch_bounds__(256) void embed_kernel(const int* __restrict__ idx,
                                                    const float* __restrict__ emb,
                                                    float* __restrict__ tokEmb) {
  int t = blockIdx.x;
  int row = idx[t];
  const float4* src = (const float4*)(emb + (size_t)row * HH);
  float4* dst = (float4*)(tokEmb + (size_t)t * HH);
  int i = threadIdx.x;                 // 256 threads x float4 == 1024 floats
  dst[i] = src[i];
}

// ---------------- K1: encPre[s,a] = enc @ Wa_x^T + ba  (f32 WMMA) ----------------
// 8 waves per block, one 16x16 tile per wave. 512 tiles total.
__global__ __launch_bounds__(256) void encpre_kernel(const float* __restrict__ enc,
                                                     const float* __restrict__ Wax,
                                                     const float* __restrict__ ba,
                                                     float* __restrict__ encPre) {
  int id = blockIdx.x * 8 + (threadIdx.x >> 5);  // tile id 0..511
  int tileS = id >> 4;                           // 0..31
  int tileA = id & 15;                           // 0..15
  int lane = threadIdx.x & 31;
  int lo = lane & 15, hi = lane >> 4;
  const float* arow = enc + (size_t)(tileS * 16 + lo) * HH;   // A: m = lo
  const float* brow = Wax + (size_t)(tileA * 16 + lo) * HH;   // B[k][n] = Wax[n][k], n = lo
  v8f acc = {};
  for (int k = 0; k < HH; k += 4) {
    int kk = k + 2 * hi;
    v2f a, b;
    a.x = arow[kk]; a.y = arow[kk + 1];
    b.x = brow[kk]; b.y = brow[kk + 1];
    acc = __builtin_amdgcn_wmma_f32_16x16x4_f32(false, a, false, b, (short)0, acc,
                                                false, false);
  }
  int col = tileA * 16 + lo;
  float bias = ba[col];
#pragma unroll
  for (int r = 0; r < 8; r++) {
    int m = tileS * 16 + r + 8 * hi;
    encPre[(size_t)m * AA + col] = acc[r] + bias;
  }
}

// ---------------- K2: persistent sequential recurrence (one workgroup) ----------------
__global__ __launch_bounds__(1024) void recurrence_kernel(
    const float* __restrict__ enc, const float* __restrict__ tokEmb,
    const float* __restrict__ encPre, const float* __restrict__ Wac,
    const float* __restrict__ va, const float* __restrict__ vab,
    const float* __restrict__ Wih0, const float* __restrict__ bih0,
    const float* __restrict__ Whh0, const float* __restrict__ bhh0,
    const float* __restrict__ Wih1, const float* __restrict__ bih1,
    const float* __restrict__ Whh1, const float* __restrict__ bhh1,
    const float* __restrict__ Wh, const float* __restrict__ bh,
    const float* __restrict__ vh, const float* __restrict__ vhb,
    float* __restrict__ H1all) {
  __shared__ __align__(16) float sh0[HH], sh1[HH], snh0[HH], snh1[HH];
  __shared__ __align__(16) float sx[2 * HH];
  __shared__ __align__(16) float sgi[3 * HH], sgh[3 * HH];
  __shared__ __align__(16) float se[SS], sctx[AA], stq[2 * AA];
  __shared__ float sred[32], sscal[4];

  const int tid = threadIdx.x;          // 0..1023
  const int lane = tid & 31;
  const int wave = tid >> 5;            // 0..31 (32 waves)
  const float vabv = vab[0];
  const float vhb0 = vhb[0];

  const float4* sx4   = (const float4*)sx;
  const float4* sh04  = (const float4*)sh0;
  const float4* sh14  = (const float4*)sh1;
  const float4* snh04 = (const float4*)snh0;
  const float4* snh14 = (const float4*)snh1;
  const float4* sctx4 = (const float4*)sctx;
  const float4* va4   = (const float4*)va;

  for (int i = tid; i < HH; i += 1024) { sh0[i] = 0.f; sh1[i] = 0.f; }
  __syncthreads();

  for (int t = 0; t < TT; t++) {
    // --- A: ctx = Wa_c @ h0  (256 wave-dots, 8 per wave, b128 loads) ---
    for (int i = 0; i < 8; i++) {
      int a = wave * 8 + i;
      const float4* row4 = (const float4*)(Wac + (size_t)a * HH);
      float p = 0.f;
      for (int k = lane; k < HH / 4; k += 32) p += dot4(row4[k], sh04[k]);
      p = waveSum(p);
      if (lane == 0) sctx[a] = p;
    }
    __syncthreads();

    // --- B: e[s] = vab + sum_a va[a]*tanh(encPre[s,a]+ctx[a]) ---
    for (int i = 0; i < 16; i++) {
      int s = wave * 16 + i;
      const float4* row4 = (const float4*)(encPre + (size_t)s * AA);
      float p = 0.f;
      for (int k = lane; k < AA / 4; k += 32) {
        float4 r = row4[k], c = sctx4[k], vv = va4[k];
        p += vv.x * tanhf(r.x + c.x) + vv.y * tanhf(r.y + c.y) +
             vv.z * tanhf(r.z + c.z) + vv.w * tanhf(r.w + c.w);
      }
      p = waveSum(p);
      if (lane == 0) se[s] = p + vabv;
    }
    __syncthreads();

    // --- C: softmax over e[0..511] ---
    {
      float v = (tid < SS) ? se[tid] : -3.0e38f;
      float m = waveMax(v);
      if (lane == 0) sred[wave] = m;
      __syncthreads();
      if (wave == 0) { float r = waveMax(sred[lane]); if (lane == 0) sscal[0] = r; }
      __syncthreads();
      float mx = sscal[0];
      float ex = (tid < SS) ? __expf(se[tid] - mx) : 0.f;
      if (tid < SS) se[tid] = ex;
      float s = waveSum(ex);
      if (lane == 0) sred[wave] = s;
      __syncthreads();
      if (wave == 0) { float r = waveSum(sred[lane]); if (lane == 0) sscal[1] = r; }
      __syncthreads();
    }
    const float invS = 1.0f / sscal[1];

    // --- D: a = softmax(e) @ enc ; x = [a ; tok_emb[t]]  (coalesced over h) ---
    {
      int h = tid;
      float acc = 0.f;
      for (int s = 0; s < SS; s++) acc += se[s] * enc[(size_t)s * HH + h];
      sx[h] = acc * invS;
      sx[HH + h] = tokEmb[(size_t)t * HH + h];
    }
    __syncthreads();

    // --- E: GRU0 gates (3072 wave-dots over 2048 and 1024, b128 loads) ---
    for (int i = 0; i < 96; i++) {
      int j = wave * 96 + i;
      const float4* ri4 = (const float4*)(Wih0 + (size_t)j * (2 * HH));
      const float4* rh4 = (const float4*)(Whh0 + (size_t)j * HH);
      if (i < 95) {  // prefetch next rows for this wave (global_prefetch_b8)
        __builtin_prefetch((const char*)(ri4 + (2 * HH) / 4) + lane * 256, 0, 0);
        __builtin_prefetch((const char*)(rh4 + HH / 4) + lane * 128, 0, 0);
      }
      float pi = 0.f, ph = 0.f;
      for (int k = lane; k < (2 * HH) / 4; k += 32) pi += dot4(ri4[k], sx4[k]);
      for (int k = lane; k < HH / 4; k += 32) ph += dot4(rh4[k], sh04[k]);
      pi = waveSum(pi); ph = waveSum(ph);
      if (lane == 0) { sgi[j] = pi + bih0[j]; sgh[j] = ph + bhh0[j]; }
    }
    __syncthreads();
    // --- F: GRU0 combine ---
    {
      int h = tid;
      float r = sigmf(sgi[h] + sgh[h]);
      float z = sigmf(sgi[HH + h] + sgh[HH + h]);
      float n = tanhf(sgi[2 * HH + h] + r * sgh[2 * HH + h]);
      snh0[h] = (1.f - z) * n + z * sh0[h];
    }
    __syncthreads();

    // --- G: GRU1 gates + combine ---
    for (int i = 0; i < 96; i++) {
      int j = wave * 96 + i;
      const float4* ri4 = (const float4*)(Wih1 + (size_t)j * HH);
      const float4* rh4 = (const float4*)(Whh1 + (size_t)j * HH);
      if (i < 95) {
        __builtin_prefetch((const char*)(ri4 + HH / 4) + lane * 128, 0, 0);
        __builtin_prefetch((const char*)(rh4 + HH / 4) + lane * 128, 0, 0);
      }
      float pi = 0.f, ph = 0.f;
      for (int k = lane; k < HH / 4; k += 32) {
        pi += dot4(ri4[k], snh04[k]);
        ph += dot4(rh4[k], sh14[k]);
      }
      pi = waveSum(pi); ph = waveSum(ph);
      if (lane == 0) { sgi[j] = pi + bih1[j]; sgh[j] = ph + bhh1[j]; }
    }
    __syncthreads();
    {
      int h = tid;
      float r = sigmf(sgi[h] + sgh[h]);
      float z = sigmf(sgi[HH + h] + sgh[HH + h]);
      float n = tanhf(sgi[2 * HH + h] + r * sgh[2 * HH + h]);
      snh1[h] = (1.f - z) * n + z * sh1[h];
    }
    __syncthreads();

    // --- H: hidden attention (layer 0) over {nh0, nh1} ---
    for (int i = 0; i < 16; i++) {
      int idx = wave * 16 + i;            // 0..511
      int a = idx & (AA - 1);
      const float4* hv4 = (idx >> 8) ? snh14 : snh04;
      const float4* row4 = (const float4*)(Wh + (size_t)a * HH);  // layer 0
      float p = 0.f;
      for (int k = lane; k < HH / 4; k += 32) p += dot4(row4[k], hv4[k]);
      p = waveSum(p);
      if (lane == 0) stq[idx] = vh[a] * tanhf(p + bh[a]);
    }
    __syncthreads();
    {
      float p0 = (tid < AA) ? stq[tid] : 0.f;
      p0 = waveSum(p0);
      if (lane == 0) sred[wave] = p0;
      __syncthreads();
      if (wave == 0) { float r = waveSum(sred[lane]); if (lane == 0) sscal[2] = r + vhb0; }
      __syncthreads();
      float p1 = (tid < AA) ? stq[AA + tid] : 0.f;
      p1 = waveSum(p1);
      if (lane == 0) sred[wave] = p1;
      __syncthreads();
      if (wave == 0) { float r = waveSum(sred[lane]); if (lane == 0) sscal[3] = r + vhb0; }
      __syncthreads();
    }
    // --- I: 2-way softmax blend; h1n == nh1 (layer-1 attn is identity) ---
    {
      float e0 = sscal[2], e1 = sscal[3];
      float m = fmaxf(e0, e1);
      float x0 = __expf(e0 - m), x1 = __expf(e1 - m);
      float w0 = x0 / (x0 + x1), w1 = x1 / (x0 + x1);
      int h = tid;
      float nh1v = snh1[h];
      H1all[(size_t)t * HH + h] = nh1v;
      sh0[h] = w0 * snh0[h] + w1 * nh1v;
      sh1[h] = nh1v;
    }
    __syncthreads();
  }
}

// ---------------- K3: fc1[t,j] = relu(Wf1 @ h1n[t] + bf1) ----------------
__global__ __launch_bounds__(256) void head_kernel(const float* __restrict__ H1all,
                                                   const float* __restrict__ Wf1,
                                                   const float* __restrict__ bf1,
                                                   float* __restrict__ fc1all) {
  int t = blockIdx.x;
  int lane = threadIdx.x & 31, wave = threadIdx.x >> 5;  // 8 waves
  const float4* hrow4 = (const float4*)(H1all + (size_t)t * HH);
#pragma unroll
  for (int i = 0; i < 4; i++) {
    int j = wave * 4 + i;
    const float4* w4 = (const float4*)(Wf1 + (size_t)j * HH);
    float p = 0.f;
    for (int k = lane; k < HH / 4; k += 32) p += dot4(w4[k], hrow4[k]);
    p = waveSum(p);
    if (lane == 0) fc1all[t * 32 + j] = fmaxf(p + bf1[j], 0.f);
  }
}

// ---------------- K4: logits[t,v] = fc1 @ Wf2^T + bf2  (f32 WMMA; M=512,N=50257,K=32) ----
// 8 waves per block, one 16x16 tile per wave.
__global__ __launch_bounds__(256) void logits_kernel(const float* __restrict__ fc1all,
                                                     const float* __restrict__ Wf2,
                                                     const float* __restrict__ bf2,
                                                     float* __restrict__ out) {
  int tn = blockIdx.x * 8 + (threadIdx.x >> 5);  // vocab tile 0..3141 (wave-uniform)
  if (tn >= NV_TILES) return;                    // whole wave exits together (EXEC stays full)
  int tm = blockIdx.y;                           // time tile 0..31
  int lane = threadIdx.x & 31;
  int lo = lane & 15, hi = lane >> 4;
  const float* arow = fc1all + (size_t)(tm * 16 + lo) * 32;
  int vrow = tn * 16 + lo;
  int vclamp = (vrow < VV) ? vrow : (VV - 1);
  float bmask = (vrow < VV) ? 1.f : 0.f;
  const float* brow = Wf2 + (size_t)vclamp * 32;
  v8f acc = {};
#pragma unroll
  for (int k = 0; k < 32; k += 4) {
    int kk = k + 2 * hi;
    v2f a, b;
    a.x = arow[kk];          a.y = arow[kk + 1];
    b.x = brow[kk] * bmask;  b.y = brow[kk + 1] * bmask;
    acc = __builtin_amdgcn_wmma_f32_16x16x4_f32(false, a, false, b, (short)0, acc,
                                                false, false);
  }
  int col = tn * 16 + lo;
  if (col < VV) {
    float bias = bf2[col];
#pragma unroll
    for (int r = 0; r < 8; r++) {
      int m = tm * 16 + r + 8 * hi;
      out[(size_t)m * VV + col] = acc[r] + bias;
    }
  }
}

// ---------------- K5: in-place row softmax over vocab ----------------
__global__ __launch_bounds__(1024) void softmax_kernel(float* __restrict__ out) {
  __shared__ float sred[32], ss[2];
  int t = blockIdx.x;
  float* row = out + (size_t)t * VV;
  int tid = threadIdx.x, lane = tid & 31, wave = tid >> 5;

  float m = -3.0e38f;
  for (int i = tid; i < VV; i += 1024) m = fmaxf(m, row[i]);
  m = waveMax(m);
  if (lane == 0) sred[wave] = m;
  __syncthreads();
  if (wave == 0) { float r = waveMax(sred[lane]); if (lane == 0) ss[0] = r; }
  __syncthreads();
  m = ss[0];

  float s = 0.f;
  for (int i = tid; i < VV; i += 1024) {
    float p = __expf(row[i] - m);
    row[i] = p;
    s += p;
  }
  s = waveSum(s);
  if (lane == 0) sred[wave] = s;
  __syncthreads();
  if (wave == 0) { float r = waveSum(sred[lane]); if (lane == 0) ss[1] = r; }
  __syncthreads();
  float inv = 1.0f / ss[1];
  for (int i = tid; i < VV; i += 1024) row[i] *= inv;
}

// ---------------- launch ----------------
extern "C" void kernel_launch(void* const* d_in, const int* in_sizes, int n_in,
                              void* d_out, int out_size, void* d_ws, size_t ws_size,
                              hipStream_t stream) {
  (void)in_sizes; (void)n_in; (void)out_size; (void)ws_size;
  const int*   dec  = (const int*)  d_in[0];
  const float* enc  = (const float*)d_in[1];
  const float* emb  = (const float*)d_in[2];
  const float* Wax  = (const float*)d_in[3];
  const float* Wac  = (const float*)d_in[4];
  const float* ba   = (const float*)d_in[5];
  const float* va   = (const float*)d_in[6];
  const float* vab  = (const float*)d_in[7];
  const float* Wih0 = (const float*)d_in[8];
  const float* bih0 = (const float*)d_in[9];
  const float* Whh0 = (const float*)d_in[10];
  const float* bhh0 = (const float*)d_in[11];
  const float* Wih1 = (const float*)d_in[12];
  const float* bih1 = (const float*)d_in[13];
  const float* Whh1 = (const float*)d_in[14];
  const float* bhh1 = (const float*)d_in[15];
  const float* Wh   = (const float*)d_in[16];
  const float* bh   = (const float*)d_in[17];
  const float* vh   = (const float*)d_in[18];
  const float* vhb  = (const float*)d_in[19];
  const float* Wf1  = (const float*)d_in[20];
  const float* bf1  = (const float*)d_in[21];
  const float* Wf2  = (const float*)d_in[22];
  const float* bf2  = (const float*)d_in[23];
  float* out = (float*)d_out;

  // workspace layout (floats): ~4.6 MB total
  float* ws     = (float*)d_ws;
  float* tokEmb = ws;                      // 512*1024
  float* encPre = tokEmb + TT * HH;        // 512*256
  float* H1all  = encPre + SS * AA;        // 512*1024
  float* fc1all = H1all + TT * HH;         // 512*32

  embed_kernel<<<TT, 256, 0, stream>>>(dec, emb, tokEmb);
  encpre_kernel<<<512 / 8, 256, 0, stream>>>(enc, Wax, ba, encPre);
  recurrence_kernel<<<1, 1024, 0, stream>>>(enc, tokEmb, encPre, Wac, va, vab,
                                            Wih0, bih0, Whh0, bhh0,
                                            Wih1, bih1, Whh1, bhh1,
                                            Wh, bh, vh, vhb, H1all);
  head_kernel<<<TT, 256, 0, stream>>>(H1all, Wf1, bf1, fc1all);
  logits_kernel<<<dim3((NV_TILES + 7) / 8, TT / 16), 256, 0, stream>>>(fc1all, Wf2, bf2, out);
  softmax_kernel<<<TT, 1024, 0, stream>>>(out);
}